// SimpleGenerator_40149354283201
// MI455X (gfx1250) — compile-verified
//
#include <hip/hip_runtime.h>
#include <hip/hip_bf16.h>

// ---------------------------------------------------------------------------
// Problem constants (match reference)
// ---------------------------------------------------------------------------
#define N_MIRNA   100000
#define N_DISEASE 2048
#define N_TARGET  50000
#define N_EDGES   1000000
#define ITEM_COUNT 2048
#define FEAT      256
#define OUT_FEAT  128
#define SIZE_ROWS 16384

typedef __attribute__((ext_vector_type(2))) float v2f;
typedef __attribute__((ext_vector_type(8))) float v8f;

// ---------------------------------------------------------------------------
// Generic fp32 WMMA GEMM:  C = act( rowscale(A_cat) @ W + bias )
//   A_cat(m,k) = (!CONCAT || k < Ksplit) ? A[m*lda+k] : A2[m*lda2 + (k-Ksplit)]
//   rowscale   = rsqrt(max(deg[m],1)) if USE_DEG
//
// Tiling: each block covers 64 rows; each wave owns a 64x16 output tile
// (4 x v8f accumulators). Per K-step of 4:
//   - one B fragment from global (b32 + b32 offset:N*4 off one pointer),
//     reused by 4 WMMAs; pointer bumped by a constant (no 64-bit muls)
//   - four A fragments from LDS (ds_load_b64, stride-132 -> conflict free)
// N / ACT / USE_DEG / CONCAT are compile-time to keep the K-loop lean.
// ---------------------------------------------------------------------------
#define KC 128
#define MR 4                 // 16-row subtiles per wave
#define MTILE (16 * MR)      // 64 rows per block
#define LDS_STRIDE 132

template<int N_, int ACT, bool USE_DEG, bool CONCAT>
__global__ void __launch_bounds__(256) gemm_wmma_f32(
    const float* __restrict__ A, int lda,
    const float* __restrict__ A2, int lda2, int Ksplit,
    const float* __restrict__ deg,          // row-scale source (USE_DEG)
    const float* __restrict__ W, int K, int M,
    const float* __restrict__ bias,         // optional, length N_
    float* __restrict__ C)                  // ldc == N_
{
    __shared__ float As[MTILE * LDS_STRIDE];   // 64 x 132 floats = 33.8 KB

    const int tid    = threadIdx.x;
    const int lane   = tid & 31;
    const int wave   = tid >> 5;
    const int nwaves = blockDim.x >> 5;
    const int rowBase = blockIdx.x * MTILE;
    const int colBase = (blockIdx.y * nwaves + wave) * 16;
    const int lane15  = lane & 15;
    const int kHi     = (lane >> 4) * 2;   // 0 for lanes 0-15, 2 for lanes 16-31

    v8f acc[MR];
    #pragma unroll
    for (int r = 0; r < MR; ++r) acc[r] = (v8f){};

    for (int kc = 0; kc < K; kc += KC) {
        const int chunk = (K - kc < KC) ? (K - kc) : KC;   // always a multiple of 8

        // --- cooperative, coalesced stage of the 64-row A slab into LDS
        for (int idx = tid; idx < MTILE * chunk; idx += blockDim.x) {
            const int r  = idx / chunk;
            const int c  = idx - r * chunk;
            const int gk = kc + c;
            int gm = rowBase + r;
            if (gm >= M) gm = M - 1;                       // clamp (stores guarded)
            float v;
            if (CONCAT)
                v = (gk < Ksplit) ? A[(long)gm * lda + gk]
                                  : A2[(long)gm * lda2 + (gk - Ksplit)];
            else
                v = A[(long)gm * lda + gk];
            if (USE_DEG) v *= rsqrtf(fmaxf(deg[gm], 1.0f));
            As[r * LDS_STRIDE + c] = v;
            // prefetch next K-slab of A while we are here (global_prefetch_b8)
            if (kc + KC < K) {
                const int pk = gk + KC;
                const float* pp;
                if (CONCAT)
                    pp = (pk < Ksplit) ? &A[(long)gm * lda + pk]
                                       : &A2[(long)gm * lda2 + (pk - Ksplit)];
                else
                    pp = &A[(long)gm * lda + pk];
                __builtin_prefetch(pp, 0, 1);
            }
        }
        __syncthreads();

        // --- WMMA K-loop over the staged slab
        const float* __restrict__ wp = W + (long)(kc + kHi) * N_ + colBase + lane15;
        #pragma unroll 2
        for (int kk = 0; kk < chunk; kk += 4) {
            // B fragment: wp[0] and wp[N_] (immediate offset), then constant bump
            v2f b;
            b.x = wp[0];
            b.y = wp[N_];
            // 4 row-subtiles reuse the same B fragment
            #pragma unroll
            for (int r = 0; r < MR; ++r) {
                const v2f a = *reinterpret_cast<const v2f*>(
                    &As[(r * 16 + lane15) * LDS_STRIDE + kk + kHi]);
                acc[r] = __builtin_amdgcn_wmma_f32_16x16x4_f32(
                             false, a, false, b, (short)0, acc[r], false, false);
            }
            wp += 4 * N_;
        }
        __syncthreads();
    }

    // --- epilogue: D layout = 8 VGPRs, M split at lane 16 (ISA 7.12.2)
    const int n  = colBase + lane15;
    const float bv = bias ? bias[n] : 0.0f;
    #pragma unroll
    for (int r = 0; r < MR; ++r) {
        const int m0 = rowBase + r * 16 + (lane >> 4) * 8;
        #pragma unroll
        for (int i = 0; i < 8; ++i) {
            float v = acc[r][i] + bv;
            if (ACT == 1)      v = fmaxf(v, 0.0f);
            else if (ACT == 2) v = 1.0f / (1.0f + __expf(-v));
            if (m0 + i < M) C[(long)(m0 + i) * N_ + n] = v;
        }
    }
}

// ---------------------------------------------------------------------------
// Degree counting for both edge lists (float counters for later rsqrt)
// ---------------------------------------------------------------------------
__global__ void __launch_bounds__(256) deg_kernel(
    const int* __restrict__ esd, const int* __restrict__ edd,
    const int* __restrict__ est, const int* __restrict__ edt,
    float* __restrict__ dod, float* __restrict__ dot,
    float* __restrict__ did, float* __restrict__ dit, int n)
{
    const int i = blockIdx.x * blockDim.x + threadIdx.x;
    if (i < n) {
        atomicAdd(&dod[esd[i]], 1.0f);
        atomicAdd(&did[edd[i]], 1.0f);
        atomicAdd(&dot[est[i]], 1.0f);
        atomicAdd(&dit[edt[i]], 1.0f);
    }
}

// ---------------------------------------------------------------------------
// Edge scatter: mirna[dst] += x[src] * rsqrt(max(deg_in[dst],1))
// (deg_in scale folded into each contribution — exact by linearity)
// 2 edges per 256-thread block; 128 lanes cover the 128 features.
// ---------------------------------------------------------------------------
__global__ void __launch_bounds__(256) scatter_kernel(
    const float* __restrict__ x,
    const int* __restrict__ esrc, const int* __restrict__ edst,
    const float* __restrict__ deg_in,
    float* __restrict__ mirna, int nEdges)
{
    const int t = threadIdx.x & 127;
    const int e = blockIdx.x * 2 + (threadIdx.x >> 7);
    if (e >= nEdges) return;
    const int s = esrc[e];
    const int d = edst[e];
    const float rs = rsqrtf(fmaxf(deg_in[d], 1.0f));
    atomicAdd(&mirna[(long)d * OUT_FEAT + t], x[(long)s * OUT_FEAT + t] * rs);
}

// ---------------------------------------------------------------------------
// Bias + L1 row-normalize of window rows -> fake_embedding
// ---------------------------------------------------------------------------
__global__ void __launch_bounds__(128) norm_kernel(
    const float* __restrict__ mirna,
    const float* __restrict__ bd, const float* __restrict__ bt,
    const int* __restrict__ leftIndex,
    float* __restrict__ fake)
{
    __shared__ float red[128];
    const int r = blockIdx.x;
    const int t = threadIdx.x;
    const int row = leftIndex[0] + r;
    const float v = mirna[(long)row * OUT_FEAT + t] + bd[t] + bt[t];
    red[t] = fabsf(v);
    __syncthreads();
    for (int s = 64; s > 0; s >>= 1) {
        if (t < s) red[t] += red[t + s];
        __syncthreads();
    }
    const float total = fmaxf(red[0], 1e-12f);
    fake[(long)r * OUT_FEAT + t] = v / total;
}

// ---------------------------------------------------------------------------
// Host side
// ---------------------------------------------------------------------------
extern "C" void kernel_launch(void* const* d_in, const int* in_sizes, int n_in,
                              void* d_out, int out_size, void* d_ws, size_t ws_size,
                              hipStream_t stream)
{
    const float* h_disease = (const float*)d_in[0];
    const float* h_target  = (const float*)d_in[1];
    const float* Adj       = (const float*)d_in[2];
    const int*   esrc_dm   = (const int*)  d_in[3];
    const int*   edst_dm   = (const int*)  d_in[4];
    const int*   esrc_tm   = (const int*)  d_in[5];
    const int*   edst_tm   = (const int*)  d_in[6];
    const float* Wd        = (const float*)d_in[7];
    const float* bd        = (const float*)d_in[8];
    const float* Wt        = (const float*)d_in[9];
    const float* bt        = (const float*)d_in[10];
    const float* W1        = (const float*)d_in[11];
    const float* b1        = (const float*)d_in[12];
    const float* W2        = (const float*)d_in[13];
    const float* b2        = (const float*)d_in[14];
    const float* W3        = (const float*)d_in[15];
    const float* b3        = (const float*)d_in[16];
    const int*   leftIndex = (const int*)  d_in[18];

    // Workspace layout (floats)
    float* ws        = (float*)d_ws;
    float* deg_out_d = ws;                                  // 2048
    float* deg_out_t = deg_out_d + N_DISEASE;               // 50000
    float* deg_in_d  = deg_out_t + N_TARGET;                // 100000
    float* deg_in_t  = deg_in_d  + N_MIRNA;                 // 100000
    float* mirna     = deg_in_t  + N_MIRNA;                 // 100000*128
    float* xd        = mirna + (size_t)N_MIRNA * OUT_FEAT;  // 2048*128
    float* xt        = xd    + (size_t)N_DISEASE * OUT_FEAT;// 50000*128
    float* h1        = xt    + (size_t)N_TARGET * OUT_FEAT; // 16384*128
    float* h2        = h1    + (size_t)SIZE_ROWS * 128;     // 16384*64

    // Zero counters + aggregation buffer (capturable async memset)
    const size_t zeroFloats = (size_t)N_DISEASE + N_TARGET + 2 * (size_t)N_MIRNA
                            + (size_t)N_MIRNA * OUT_FEAT;
    hipMemsetAsync(d_ws, 0, zeroFloats * sizeof(float), stream);

    // 1) degrees for both graphs
    deg_kernel<<<(N_EDGES + 255) / 256, 256, 0, stream>>>(
        esrc_dm, edst_dm, esrc_tm, edst_tm,
        deg_out_d, deg_out_t, deg_in_d, deg_in_t, N_EDGES);

    // 2) xd = (h_disease * rsqrt(deg_out_d)) @ Wd      [2048x256 @ 256x128]
    gemm_wmma_f32<OUT_FEAT, 0, true, false>
        <<<dim3((N_DISEASE + MTILE - 1) / MTILE, 1), 256, 0, stream>>>(
        h_disease, FEAT, nullptr, 0, FEAT, deg_out_d,
        Wd, FEAT, N_DISEASE, nullptr, xd);

    // 3) xt = (h_target * rsqrt(deg_out_t)) @ Wt       [50000x256 @ 256x128]
    gemm_wmma_f32<OUT_FEAT, 0, true, false>
        <<<dim3((N_TARGET + MTILE - 1) / MTILE, 1), 256, 0, stream>>>(
        h_target, FEAT, nullptr, 0, FEAT, deg_out_t,
        Wt, FEAT, N_TARGET, nullptr, xt);

    // 4) edge scatter with fused rsqrt(deg_in) (both graphs into one buffer)
    scatter_kernel<<<(N_EDGES + 1) / 2, 256, 0, stream>>>(
        xd, esrc_dm, edst_dm, deg_in_d, mirna, N_EDGES);
    scatter_kernel<<<(N_EDGES + 1) / 2, 256, 0, stream>>>(
        xt, esrc_tm, edst_tm, deg_in_t, mirna, N_EDGES);

    // 5) bias + L1 normalize window -> fake_embedding (first output)
    float* fake = (float*)d_out;                        // 16384*128
    float* outp = fake + (size_t)SIZE_ROWS * OUT_FEAT;  // 16384*2048
    norm_kernel<<<SIZE_ROWS, 128, 0, stream>>>(mirna, bd, bt, leftIndex, fake);

    // 6) h1 = relu([Adj | fake] @ W1 + b1)   [16384x2176 @ 2176x128]
    gemm_wmma_f32<128, 1, false, true>
        <<<dim3(SIZE_ROWS / MTILE, 1), 256, 0, stream>>>(
        Adj, ITEM_COUNT, fake, OUT_FEAT, ITEM_COUNT,
        nullptr, W1, ITEM_COUNT + OUT_FEAT, SIZE_ROWS, b1, h1);

    // 7) h2 = relu(h1 @ W2 + b2)             [16384x128 @ 128x64]
    gemm_wmma_f32<64, 1, false, false>
        <<<dim3(SIZE_ROWS / MTILE, 1), 128, 0, stream>>>(
        h1, 128, nullptr, 0, 128,
        nullptr, W2, 128, SIZE_ROWS, b2, h2);

    // 8) output = sigmoid(h2 @ W3 + b3)      [16384x64 @ 64x2048]
    gemm_wmma_f32<2048, 2, false, false>
        <<<dim3(SIZE_ROWS / MTILE, 2048 / (16 * 8)), 256, 0, stream>>>(
        h2, 64, nullptr, 0, 64,
        nullptr, W3, 64, SIZE_ROWS, b3, outp);
}